// OurSE3Transformer_32152125178060
// MI455X (gfx1250) — compile-verified
//
#include <hip/hip_runtime.h>
#include <math.h>

// ---------------- problem constants ----------------
#define E_N 50000
#define N_N 10000
#define CMID 16
#define EPS_F 1e-8f

typedef __attribute__((ext_vector_type(2))) float v2f;
typedef __attribute__((ext_vector_type(8))) float v8f;

// D = A(16x4) * B(4x16) + C(16x16), full f32 (reference precision).
__device__ __forceinline__ v8f wmma4(v2f a, v2f b, v8f c) {
  return __builtin_amdgcn_wmma_f32_16x16x4_f32(false, a, false, b, (short)0, c,
                                               false, false);
}

// ---------------- geometry ----------------
__global__ void geomK(const float* __restrict__ x, const int* __restrict__ src,
                      const int* __restrict__ dst, const float* __restrict__ ef,
                      float* __restrict__ featg, float* __restrict__ rhatg) {
  int e = blockIdx.x * blockDim.x + threadIdx.x;
  if (e >= E_N) return;
  int s = src[e], d = dst[e];
  float dx = x[d * 3 + 0] - x[s * 3 + 0];
  float dy = x[d * 3 + 1] - x[s * 3 + 1];
  float dz = x[d * 3 + 2] - x[s * 3 + 2];
  float r = sqrtf(dx * dx + dy * dy + dz * dz + EPS_F);
  float inv = 1.0f / r;
  rhatg[e * 3 + 0] = dx * inv;
  rhatg[e * 3 + 1] = dy * inv;
  rhatg[e * 3 + 2] = dz * inv;
  featg[e * 8 + 0] = r;
  featg[e * 8 + 1] = ef[e * 4 + 0];
  featg[e * 8 + 2] = ef[e * 4 + 1];
  featg[e * 8 + 3] = ef[e * 4 + 2];
  featg[e * 8 + 4] = ef[e * 4 + 3];
  featg[e * 8 + 5] = 0.0f;
  featg[e * 8 + 6] = 0.0f;
  featg[e * 8 + 7] = 0.0f;
}

__global__ void zeroK(float* __restrict__ p, int n) {
  int i = blockIdx.x * blockDim.x + threadIdx.x;
  if (i < n) p[i] = 0.0f;
}

// ---------------- node queries ----------------
__global__ void qK(const float* __restrict__ h0, const float* __restrict__ h1,
                   const float* __restrict__ wq0, const float* __restrict__ wq1,
                   float* __restrict__ q0, float* __restrict__ q1, int cin) {
  int n = blockIdx.x * blockDim.x + threadIdx.x;
  if (n >= N_N) return;
  for (int o = 0; o < 16; o++) {
    float a = 0.0f;
    for (int c = 0; c < cin; c++) a += wq0[o * cin + c] * h0[n * cin + c];
    q0[n * 16 + o] = a;
    float bx = 0.0f, by = 0.0f, bz = 0.0f;
    for (int c = 0; c < cin; c++) {
      float w = wq1[o * cin + c];
      bx += w * h1[(n * cin + c) * 3 + 0];
      by += w * h1[(n * cin + c) * 3 + 1];
      bz += w * h1[(n * cin + c) * 3 + 2];
    }
    q1[(n * 16 + o) * 3 + 0] = bx;
    q1[(n * 16 + o) * 3 + 1] = by;
    q1[(n * 16 + o) * 3 + 2] = bz;
  }
}

// ---------------- fused radial MLP (WMMA) + equivariant message ----------------
// One wave (32 lanes) per 16-edge tile. Nets in order '00','01','10','11'.
struct FusedArgs {
  const float* w1[4];  // (5 x 32) row-major
  const float* b1[4];  // (32)
  const float* w2[4];  // (32 x OD) row-major, OD = 16*cin*nJ
  const float* b2[4];  // (OD)
  const float* h0;     // (N x cin)
  const float* h1;     // (N x cin x 3)
  const int* src;
  const float* rhat;   // (E x 3)
  const float* featg;  // (E x 8), cols 5..7 zero
  float* m0;           // (E x 16)
  float* m1;           // (E x 16 x 3)
  int cin;
};

__global__ __launch_bounds__(32) void fused_radial_msg(FusedArgs A_) {
  __shared__ float sh0[16][16];      // h0[src] per edge,channel
  __shared__ float sh1[16][16][3];   // h1[src]
  __shared__ float scx[16][16][3];   // h1 x rhat
  __shared__ float su[16][16];       // rhat . h1
  __shared__ float srh[16][3];       // rhat per edge
  __shared__ float sW1[4][8][32];    // K-padded W1 for all 4 nets (zero rows 5..7)
  __shared__ float sH1n[16][32];     // hidden activations, one net at a time
  __shared__ float sRn[16][768];     // radial weights tile, one net at a time

  const int tid = threadIdx.x;
  const int lane16 = tid & 15;
  const int hi = tid >> 4;  // 0 or 1
  const int e0 = blockIdx.x * 16;
  const int cin = A_.cin;
  const int nJv[4] = {1, 1, 1, 3};

  if (tid < 16) {
    srh[tid][0] = A_.rhat[(e0 + tid) * 3 + 0];
    srh[tid][1] = A_.rhat[(e0 + tid) * 3 + 1];
    srh[tid][2] = A_.rhat[(e0 + tid) * 3 + 2];
  }
  __syncthreads();

  // Stage W1 (K rows 0..4 real, 5..7 zero) once; keeps the WMMA loops branch-free.
  for (int n = 0; n < 4; n++) {
    for (int idx = tid; idx < 5 * 32; idx += 32) {
      sW1[n][idx >> 5][idx & 31] = A_.w1[n][idx];
    }
    for (int idx = tid; idx < 3 * 32; idx += 32) {
      sW1[n][5 + (idx >> 5)][idx & 31] = 0.0f;
    }
  }

  for (int idx = tid; idx < 16 * cin; idx += 32) {
    int el = idx & 15, d = idx >> 4;
    int s = A_.src[e0 + el];
    float h0v = A_.h0[s * cin + d];
    float hx = A_.h1[(s * cin + d) * 3 + 0];
    float hy = A_.h1[(s * cin + d) * 3 + 1];
    float hz = A_.h1[(s * cin + d) * 3 + 2];
    float rx = srh[el][0], ry = srh[el][1], rz = srh[el][2];
    sh0[el][d] = h0v;
    sh1[el][d][0] = hx; sh1[el][d][1] = hy; sh1[el][d][2] = hz;
    scx[el][d][0] = hy * rz - hz * ry;   // (h x rhat)
    scx[el][d][1] = hz * rx - hx * rz;
    scx[el][d][2] = hx * ry - hy * rx;
    su[el][d] = hx * rx + hy * ry + hz * rz;
  }

  // A-layout regs for GEMM1: lane = edge (M), K = j + 2*hi + 4*s
  v2f a1[2];
  for (int s = 0; s < 2; s++)
    for (int j = 0; j < 2; j++)
      a1[s][j] = A_.featg[(e0 + lane16) * 8 + s * 4 + j + 2 * hi];

  // message accumulators: this lane owns edge lane16, channels hi*8..hi*8+7
  float m0a[8];
  float m1a[8][3];
  for (int i = 0; i < 8; i++) {
    m0a[i] = 0.0f;
    m1a[i][0] = 0.0f; m1a[i][1] = 0.0f; m1a[i][2] = 0.0f;
  }
  __syncthreads();

  for (int n = 0; n < 4; n++) {
    // ---- GEMM1: H1 = relu(feat @ W1 + b1), 16x32 ----
    for (int t = 0; t < 2; t++) {
      v8f acc = {0.f, 0.f, 0.f, 0.f, 0.f, 0.f, 0.f, 0.f};
      for (int s = 0; s < 2; s++) {
        v2f b;
        for (int j = 0; j < 2; j++)
          b[j] = sW1[n][s * 4 + j + 2 * hi][t * 16 + lane16];
        acc = wmma4(a1[s], b, acc);
      }
      float bv = A_.b1[n][t * 16 + lane16];
      for (int v = 0; v < 8; v++)
        sH1n[v + 8 * hi][t * 16 + lane16] = fmaxf(acc[v] + bv, 0.0f);
    }
    __syncthreads();

    // ---- GEMM2: R = H1 @ W2 + b2, 16 x OD ----
    const int OD = 16 * cin * nJv[n];
    v2f a2[8];
    for (int s = 0; s < 8; s++)
      for (int j = 0; j < 2; j++)
        a2[s][j] = sH1n[lane16][s * 4 + j + 2 * hi];
    const int nt = OD >> 4;
    for (int t = 0; t < nt; t++) {
      if (t + 1 < nt)  // pull next weight column tile toward L2/L0 (global_prefetch_b8)
        __builtin_prefetch(&A_.w2[n][(2 * hi) * OD + (t + 1) * 16 + lane16], 0, 1);
      v8f acc = {0.f, 0.f, 0.f, 0.f, 0.f, 0.f, 0.f, 0.f};
      for (int s = 0; s < 8; s++) {
        v2f b;
        for (int j = 0; j < 2; j++) {
          int K = s * 4 + j + 2 * hi;
          b[j] = A_.w2[n][K * OD + t * 16 + lane16];
        }
        acc = wmma4(a2[s], b, acc);
      }
      float bv = A_.b2[n][t * 16 + lane16];
      for (int v = 0; v < 8; v++)
        sRn[v + 8 * hi][t * 16 + lane16] = acc[v] + bv;
    }
    __syncthreads();

    // ---- contraction with basis into messages (VALU) ----
    {
      const int e = lane16;
      const float rx = srh[e][0], ry = srh[e][1], rz = srh[e][2];
      for (int c8 = 0; c8 < 8; c8++) {
        int c = hi * 8 + c8;
        if (n == 0) {  // Q00 = 1, deg0->deg0
          float s = 0.0f;
          for (int d = 0; d < cin; d++) s += sRn[e][c * cin + d] * sh0[e][d];
          m0a[c8] += s;
        } else if (n == 1) {  // Q01 = rhat, deg0->deg1
          float s = 0.0f;
          for (int d = 0; d < cin; d++) s += sRn[e][c * cin + d] * sh0[e][d];
          m1a[c8][0] += s * rx;
          m1a[c8][1] += s * ry;
          m1a[c8][2] += s * rz;
        } else if (n == 2) {  // Q10 = rhat^T, deg1->deg0
          float s = 0.0f;
          for (int d = 0; d < cin; d++) s += sRn[e][c * cin + d] * su[e][d];
          m0a[c8] += s;
        } else {  // Q11: identity / cross / rank-2 traceless, deg1->deg1
          float ax = 0.0f, ay = 0.0f, az = 0.0f;
          for (int d = 0; d < cin; d++) {
            int b = (c * cin + d) * 3;
            float R0 = sRn[e][b + 0], R1 = sRn[e][b + 1], R2 = sRn[e][b + 2];
            float hx = sh1[e][d][0], hy = sh1[e][d][1], hz = sh1[e][d][2];
            float cx = scx[e][d][0], cy = scx[e][d][1], cz = scx[e][d][2];
            float uv = su[e][d];
            ax += R0 * hx + R1 * cx + R2 * (rx * uv - hx * (1.0f / 3.0f));
            ay += R0 * hy + R1 * cy + R2 * (ry * uv - hy * (1.0f / 3.0f));
            az += R0 * hz + R1 * cz + R2 * (rz * uv - hz * (1.0f / 3.0f));
          }
          m1a[c8][0] += ax;
          m1a[c8][1] += ay;
          m1a[c8][2] += az;
        }
      }
    }
    __syncthreads();
  }

  // write messages
  {
    int eg = e0 + lane16;
    for (int c8 = 0; c8 < 8; c8++) {
      int c = hi * 8 + c8;
      A_.m0[eg * 16 + c] = m0a[c8];
      A_.m1[(eg * 16 + c) * 3 + 0] = m1a[c8][0];
      A_.m1[(eg * 16 + c) * 3 + 1] = m1a[c8][1];
      A_.m1[(eg * 16 + c) * 3 + 2] = m1a[c8][2];
    }
  }
}

// ---------------- attention ----------------
__device__ __forceinline__ unsigned f2key(float x) {
  unsigned u = __float_as_uint(x);
  return (u & 0x80000000u) ? ~u : (u | 0x80000000u);
}
__device__ __forceinline__ float key2f(unsigned k) {
  unsigned u = (k & 0x80000000u) ? (k ^ 0x80000000u) : ~k;
  return __uint_as_float(u);
}

__global__ void logitsK(const float* __restrict__ q0, const float* __restrict__ q1,
                        const int* __restrict__ dst, const float* __restrict__ k0,
                        const float* __restrict__ k1, float* __restrict__ lg) {
  int e = blockIdx.x * blockDim.x + threadIdx.x;
  if (e >= E_N) return;
  int d = dst[e];
  float a = 0.0f;
  for (int c = 0; c < 16; c++) a += q0[d * 16 + c] * k0[e * 16 + c];
  for (int i = 0; i < 48; i++) a += q1[d * 48 + i] * k1[e * 48 + i];
  lg[e] = a * 0.125f;  // 1/sqrt(C_MID*(1+3)) = 1/8
}

__global__ void maxK(const float* __restrict__ lg, const int* __restrict__ dst,
                     unsigned* __restrict__ nmax) {
  int e = blockIdx.x * blockDim.x + threadIdx.x;
  if (e >= E_N) return;
  atomicMax(&nmax[dst[e]], f2key(lg[e]));
}

__global__ void expK(const float* __restrict__ lg, const int* __restrict__ dst,
                     const unsigned* __restrict__ nmax, float* __restrict__ eb,
                     float* __restrict__ nsum) {
  int e = blockIdx.x * blockDim.x + threadIdx.x;
  if (e >= E_N) return;
  int d = dst[e];
  float ev = expf(lg[e] - key2f(nmax[d]));
  eb[e] = ev;
  atomicAdd(&nsum[d], ev);
}

__global__ void scatterK(const float* __restrict__ eb, const int* __restrict__ dst,
                         const float* __restrict__ nsum, const float* __restrict__ v0,
                         const float* __restrict__ v1, float* __restrict__ agg0,
                         float* __restrict__ agg1) {
  int e = blockIdx.x * blockDim.x + threadIdx.x;
  if (e >= E_N) return;
  int d = dst[e];
  float alpha = eb[e] / (nsum[d] + EPS_F);
  for (int c = 0; c < 16; c++) atomicAdd(&agg0[d * 16 + c], alpha * v0[e * 16 + c]);
  for (int i = 0; i < 48; i++) atomicAdd(&agg1[d * 48 + i], alpha * v1[e * 48 + i]);
}

// ---------------- projection + equivariant norm ----------------
__global__ void projNormK(const float* __restrict__ agg0, const float* __restrict__ agg1,
                          const float* __restrict__ h0in, const float* __restrict__ h1in,
                          const float* __restrict__ wp0, const float* __restrict__ wp1,
                          const float* __restrict__ nw0, const float* __restrict__ nb0,
                          const float* __restrict__ nw1, const float* __restrict__ nb1,
                          float* __restrict__ h0out, float* __restrict__ h1out, int cin) {
  int n = blockIdx.x * blockDim.x + threadIdx.x;
  if (n >= N_N) return;
  int W = 16 + cin;
  float t0[16], t1[16][3];
  for (int o = 0; o < 16; o++) {
    float a = 0.0f;
    for (int c = 0; c < 16; c++) a += wp0[o * W + c] * agg0[n * 16 + c];
    for (int c = 0; c < cin; c++) a += wp0[o * W + 16 + c] * h0in[n * cin + c];
    t0[o] = a;
    float bx = 0.0f, by = 0.0f, bz = 0.0f;
    for (int c = 0; c < 16; c++) {
      float w = wp1[o * W + c];
      bx += w * agg1[(n * 16 + c) * 3 + 0];
      by += w * agg1[(n * 16 + c) * 3 + 1];
      bz += w * agg1[(n * 16 + c) * 3 + 2];
    }
    for (int c = 0; c < cin; c++) {
      float w = wp1[o * W + 16 + c];
      bx += w * h1in[(n * cin + c) * 3 + 0];
      by += w * h1in[(n * cin + c) * 3 + 1];
      bz += w * h1in[(n * cin + c) * 3 + 2];
    }
    t1[o][0] = bx; t1[o][1] = by; t1[o][2] = bz;
  }
  float n0[16], n1[16];
  for (int o = 0; o < 16; o++) {
    n0[o] = sqrtf(t0[o] * t0[o] + EPS_F);
    n1[o] = sqrtf(t1[o][0] * t1[o][0] + t1[o][1] * t1[o][1] + t1[o][2] * t1[o][2] + EPS_F);
  }
  for (int o = 0; o < 16; o++) {
    float s0 = nb0[o];
    for (int c = 0; c < 16; c++) s0 += n0[c] * nw0[c * 16 + o];
    s0 = fmaxf(s0, 0.0f);
    h0out[n * 16 + o] = s0 * t0[o] / n0[o];
    float s1 = nb1[o];
    for (int c = 0; c < 16; c++) s1 += n1[c] * nw1[c * 16 + o];
    s1 = fmaxf(s1, 0.0f);
    for (int p = 0; p < 3; p++)
      h1out[(n * 16 + o) * 3 + p] = s1 * t1[o][p] / n1[o];
  }
}

// ---------------- final output conv (small, VALU) ----------------
__global__ void convK(const float* __restrict__ featg, const float* __restrict__ rhatg,
                      const int* __restrict__ src, const int* __restrict__ dst,
                      const float* __restrict__ h0, const float* __restrict__ h1,
                      const float* __restrict__ w1a, const float* __restrict__ b1a,
                      const float* __restrict__ w2a, const float* __restrict__ b2a,
                      const float* __restrict__ w1b, const float* __restrict__ b1b,
                      const float* __restrict__ w2b, const float* __restrict__ b2b,
                      float* __restrict__ aggF) {
  int e = blockIdx.x * blockDim.x + threadIdx.x;
  if (e >= E_N) return;
  float f[5];
  for (int k = 0; k < 5; k++) f[k] = featg[e * 8 + k];
  float rx = rhatg[e * 3 + 0], ry = rhatg[e * 3 + 1], rz = rhatg[e * 3 + 2];
  int s = src[e], d = dst[e];
  float m0 = 0.0f, m1 = 0.0f, m2 = 0.0f;
  float H[32];
  for (int o = 0; o < 32; o++) {
    float a = b1a[o];
    for (int k = 0; k < 5; k++) a += f[k] * w1a[k * 32 + o];
    H[o] = fmaxf(a, 0.0f);
  }
  for (int c = 0; c < 16; c++) {
    float R = b2a[c];
    for (int h = 0; h < 32; h++) R += H[h] * w2a[h * 16 + c];
    float t = R * h0[s * 16 + c];
    m0 += t * rx; m1 += t * ry; m2 += t * rz;
  }
  for (int o = 0; o < 32; o++) {
    float a = b1b[o];
    for (int k = 0; k < 5; k++) a += f[k] * w1b[k * 32 + o];
    H[o] = fmaxf(a, 0.0f);
  }
  for (int c = 0; c < 16; c++) {
    float R0 = b2b[c * 3 + 0], R1 = b2b[c * 3 + 1], R2 = b2b[c * 3 + 2];
    for (int h = 0; h < 32; h++) {
      float Hv = H[h];
      R0 += Hv * w2b[h * 48 + c * 3 + 0];
      R1 += Hv * w2b[h * 48 + c * 3 + 1];
      R2 += Hv * w2b[h * 48 + c * 3 + 2];
    }
    float hx = h1[(s * 16 + c) * 3 + 0];
    float hy = h1[(s * 16 + c) * 3 + 1];
    float hz = h1[(s * 16 + c) * 3 + 2];
    float u = hx * rx + hy * ry + hz * rz;
    float cx = hy * rz - hz * ry, cy = hz * rx - hx * rz, cz = hx * ry - hy * rx;
    m0 += R0 * hx + R1 * cx + R2 * (rx * u - hx * (1.0f / 3.0f));
    m1 += R0 * hy + R1 * cy + R2 * (ry * u - hy * (1.0f / 3.0f));
    m2 += R0 * hz + R1 * cz + R2 * (rz * u - hz * (1.0f / 3.0f));
  }
  atomicAdd(&aggF[d * 3 + 0], m0);
  atomicAdd(&aggF[d * 3 + 1], m1);
  atomicAdd(&aggF[d * 3 + 2], m2);
}

__global__ void outK(const float* __restrict__ aggF, const float* __restrict__ h1,
                     const float* __restrict__ wself, const float* __restrict__ scalar,
                     float* __restrict__ out) {
  int n = blockIdx.x * blockDim.x + threadIdx.x;
  if (n >= N_N) return;
  float sc = *scalar;
  for (int p = 0; p < 3; p++) {
    float a = aggF[n * 3 + p];
    for (int c = 0; c < 16; c++) a += wself[c] * h1[(n * 16 + c) * 3 + p];
    out[n * 3 + p] = a * sc;
  }
}

// ---------------- host launcher ----------------
extern "C" void kernel_launch(void* const* d_in, const int* in_sizes, int n_in,
                              void* d_out, int out_size, void* d_ws, size_t ws_size,
                              hipStream_t stream) {
  (void)in_sizes; (void)out_size; (void)ws_size;
  if (n_in < 96) return;  // parameter-leaf layout assumption: 96 leaves

  const float* x = (const float*)d_in[0];
  const float* f0 = (const float*)d_in[1];
  const float* f1 = (const float*)d_in[2];
  const int* esrc = (const int*)d_in[3];
  const int* edst = (const int*)d_in[4];
  const float* efeat = (const float*)d_in[5];
  auto F = [&](int i) { return (const float*)d_in[i]; };

  // Leaf map (recursive dict-insertion order):
  // per layer L (base = 6 + 40*L):
  //   rk nets 00,01,10,11: +i*4 -> l1.w, l1.b, l2.w, l2.b   (i=0..3)
  //   rv nets:             +16+i*4 -> same
  //   +32 wq.0, +33 wq.1, +34 wproj.0, +35 wproj.1
  //   +36 norm0.w, +37 norm0.b, +38 norm1.w, +39 norm1.b
  // conv (86..): r01 {w1,b1,w2,b2}, r11 {w1,b1,w2,b2}, wself(94), scalar(95)

  float* ws = (float*)d_ws;
  float* featg = ws;                    // E*8
  float* rhatg = featg + E_N * 8;       // E*3
  float* h0a = rhatg + E_N * 3;         // N*16
  float* h1a = h0a + N_N * 16;          // N*48
  float* h0b = h1a + N_N * 48;
  float* h1b = h0b + N_N * 16;
  float* q0 = h1b + N_N * 48;           // N*16
  float* q1 = q0 + N_N * 16;            // N*48
  float* k0 = q1 + N_N * 48;            // E*16
  float* k1 = k0 + E_N * 16;            // E*48
  float* v0 = k1 + E_N * 48;            // E*16
  float* v1 = v0 + E_N * 16;            // E*48
  float* lg = v1 + E_N * 48;            // E
  float* eb = lg + E_N;                 // E
  unsigned* nmax = (unsigned*)(eb + E_N);  // N
  float* nsum = (float*)(nmax + N_N);   // N
  float* agg0 = nsum + N_N;             // N*16
  float* agg1 = agg0 + N_N * 16;        // N*48
  float* aggF = agg1 + N_N * 48;        // N*3

  dim3 bE(256), gE((E_N + 255) / 256);
  dim3 bN(256), gN((N_N + 255) / 256);

  geomK<<<gE, bE, 0, stream>>>(x, esrc, edst, efeat, featg, rhatg);

  const float* h0in = f0;
  const float* h1in = f1;
  float* h0out = h0a;
  float* h1out = h1a;
  for (int L = 0; L < 2; L++) {
    int cin = (L == 0) ? 1 : CMID;
    int base = 6 + L * 40;

    qK<<<gN, bN, 0, stream>>>(h0in, h1in, F(base + 32), F(base + 33), q0, q1, cin);

    FusedArgs fa{};
    fa.h0 = h0in; fa.h1 = h1in; fa.src = esrc;
    fa.rhat = rhatg; fa.featg = featg; fa.cin = cin;

    for (int i = 0; i < 4; i++) {
      fa.w1[i] = F(base + i * 4 + 0);
      fa.b1[i] = F(base + i * 4 + 1);
      fa.w2[i] = F(base + i * 4 + 2);
      fa.b2[i] = F(base + i * 4 + 3);
    }
    fa.m0 = k0; fa.m1 = k1;
    fused_radial_msg<<<dim3(E_N / 16), dim3(32), 0, stream>>>(fa);

    for (int i = 0; i < 4; i++) {
      fa.w1[i] = F(base + 16 + i * 4 + 0);
      fa.b1[i] = F(base + 16 + i * 4 + 1);
      fa.w2[i] = F(base + 16 + i * 4 + 2);
      fa.b2[i] = F(base + 16 + i * 4 + 3);
    }
    fa.m0 = v0; fa.m1 = v1;
    fused_radial_msg<<<dim3(E_N / 16), dim3(32), 0, stream>>>(fa);

    zeroK<<<gN, bN, 0, stream>>>((float*)nmax, N_N);
    zeroK<<<gN, bN, 0, stream>>>(nsum, N_N);
    zeroK<<<dim3((N_N * 16 + 255) / 256), bN, 0, stream>>>(agg0, N_N * 16);
    zeroK<<<dim3((N_N * 48 + 255) / 256), bN, 0, stream>>>(agg1, N_N * 48);

    logitsK<<<gE, bE, 0, stream>>>(q0, q1, edst, k0, k1, lg);
    maxK<<<gE, bE, 0, stream>>>(lg, edst, nmax);
    expK<<<gE, bE, 0, stream>>>(lg, edst, nmax, eb, nsum);
    scatterK<<<gE, bE, 0, stream>>>(eb, edst, nsum, v0, v1, agg0, agg1);

    projNormK<<<gN, bN, 0, stream>>>(agg0, agg1, h0in, h1in,
                                     F(base + 34), F(base + 35),
                                     F(base + 36), F(base + 37),
                                     F(base + 38), F(base + 39),
                                     h0out, h1out, cin);
    h0in = h0out; h1in = h1out;
    h0out = h0b; h1out = h1b;
  }

  zeroK<<<dim3((N_N * 3 + 255) / 256), bN, 0, stream>>>(aggF, N_N * 3);
  convK<<<gE, bE, 0, stream>>>(featg, rhatg, esrc, edst, h0in, h1in,
                               F(86), F(87), F(88), F(89),
                               F(90), F(91), F(92), F(93), aggF);
  outK<<<gN, bN, 0, stream>>>(aggF, h1in, F(94), F(95), (float*)d_out);
}